// Photodetector_42820823941643
// MI455X (gfx1250) — compile-verified
//
#include <hip/hip_runtime.h>
#include <math.h>

typedef __attribute__((ext_vector_type(2))) float v2f;
typedef __attribute__((ext_vector_type(8))) float v8f;

#define T_LEN    32768
#define B_COLS   1024
#define CHUNK    16
#define NCHUNK   (T_LEN / CHUNK)          // 2048
#define NCTILE   (B_COLS / 16)            // 64
#define NTILES   (NCHUNK * NCTILE)        // 131072
#define FIXTILES ((NCHUNK - 1) * NCTILE)  // 131008

// Butterworth(order=4, wn=0.25) coefficients, matching the reference's
// bilinear-transform construction. b is symmetric so the reference's
// reversed-FIR indexing equals a plain convolution.
__constant__ float g_B[5] = {0.01020948f, 0.04083792f, 0.06125689f, 0.04083792f, 0.01020948f};
__constant__ float g_A[5] = {1.0f, -1.9684277f, 1.7358607f, -0.7244708f, 0.1203896f};

// 2*Q*RESP*CUTOFF and (4*KB*TEMP*CUTOFF/RLOAD)^2
#define SHOT_COEF 6.408706252e-9f
#define VAR_TH    1.09796608e-31f

// ---- operator construction (tiny scalar recurrences, built once in LDS) ----
// L: zero-state 16x16 impulse response of y[t] = u[t] - sum a_m y[t-m].
// first_chunk variant: rows 0..4 are identity (reference init: out[0:5]=FIR,
// feedback carry starts from FIR outputs 1..4; row 0 never feeds back).
__device__ inline void build_L(float* L, bool first_chunk) {
  for (int t = 0; t < 16; ++t)
    for (int j = 0; j < 16; ++j) {
      float v;
      if (first_chunk && t <= 4) {
        v = (t == j) ? 1.0f : 0.0f;
      } else if (t < j) {
        v = 0.0f;
      } else {
        v = (t == j) ? 1.0f : 0.0f;
        for (int m = 1; m <= 4; ++m)
          if (t - m >= 0) v -= g_A[m] * L[(t - m) * 16 + j];
      }
      L[t * 16 + j] = v;
    }
}

// P: 16x4 response to the 4 carried-in outputs s[i] = y[t0-4+i].
__device__ inline void build_P(float* P) {
  for (int t = 0; t < 16; ++t)
    for (int i = 0; i < 4; ++i) {
      float v = 0.0f;
      for (int m = 1; m <= 4; ++m) {
        int tm = t - m;
        float prev = (tm >= 0) ? P[tm * 4 + i] : ((tm == i - 4) ? 1.0f : 0.0f);
        v -= g_A[m] * prev;
      }
      P[t * 4 + i] = v;
    }
}

// ---- kernel 1: per-row mean -> sigma[t] --------------------------------
__global__ void __launch_bounds__(256) sigma_kernel(const float* __restrict__ sig,
                                                    float* __restrict__ sigma) {
  __shared__ float red[8];
  for (int row = blockIdx.x; row < T_LEN; row += gridDim.x) {
    const float* p = sig + (size_t)row * B_COLS;
    float s = 0.0f;
    for (int i = threadIdx.x; i < B_COLS; i += blockDim.x) s += p[i];
    for (int off = 16; off > 0; off >>= 1) s += __shfl_down(s, off, 32);
    const int lane = threadIdx.x & 31, wv = threadIdx.x >> 5;
    if (lane == 0) red[wv] = s;
    __syncthreads();
    if (threadIdx.x == 0) {
      float tot = 0.0f;
      for (int w = 0; w < 8; ++w) tot += red[w];
      float mean = tot * (1.0f / B_COLS);
      sigma[row] = sqrtf(SHOT_COEF * mean + VAR_TH);
    }
    __syncthreads();
  }
}

// ---- kernel 2: zero-state response z = L @ u per 16x16 tile (WMMA) ------
__global__ void __launch_bounds__(256) zstate_kernel(const float* __restrict__ sig,
                                                     const float* __restrict__ noi,
                                                     const float* __restrict__ sigma,
                                                     float* __restrict__ out) {
  __shared__ float Lsh[256];
  __shared__ float L0sh[256];
  __shared__ float Ush[8][16 * 17];   // per-wave u staging, padded stride
  if (threadIdx.x == 0) { build_L(Lsh, false); build_L(L0sh, true); }
  __syncthreads();

  const int lane  = threadIdx.x & 31;
  const int wave  = threadIdx.x >> 5;
  const int col   = lane & 15;
  const int half  = lane >> 4;        // 0: lanes 0-15, 1: lanes 16-31
  const int rbase = half * 8;
  float* u = Ush[wave];

  const int wglobal = blockIdx.x * (blockDim.x >> 5) + wave;
  const int nwaves  = gridDim.x * (blockDim.x >> 5);

  for (int tile = wglobal; tile < NTILES; tile += nwaves) {   // wave-uniform
    const int chunk = tile >> 6;
    const int ctile = tile & 63;
    const int c     = ctile * 16 + col;
    const int t0    = chunk * CHUNK;

    // prefetch next grid-stride tile (global_prefetch_b8 path)
    {
      const int ntile = tile + nwaves;
      if (ntile < NTILES) {
        const int nc  = (ntile & 63) * 16 + col;
        const int nt0 = (ntile >> 6) * CHUNK;
        const size_t nidx = (size_t)(nt0 + rbase) * B_COLS + nc;
        __builtin_prefetch(&sig[nidx], 0, 3);
        __builtin_prefetch(&noi[nidx], 0, 3);
      }
    }

    // FIR of noisy input: u[t] = sum_j B[j] * (sig + sigma*noise)[t-j]
    float xn[12];
    if (chunk != 0) {
      // wave-uniform fast path: no t>=0 guard -> branch-free clause of loads
#pragma unroll
      for (int q = 0; q < 12; ++q) {
        const int t = t0 + rbase + q - 4;
        const size_t idx = (size_t)t * B_COLS + c;
        xn[q] = sig[idx] + sigma[t] * noi[idx];
      }
    } else {
#pragma unroll
      for (int q = 0; q < 12; ++q) {
        const int t = t0 + rbase + q - 4;
        float v = 0.0f;
        if (t >= 0) {
          const size_t idx = (size_t)t * B_COLS + c;
          v = sig[idx] + sigma[t] * noi[idx];
        }
        xn[q] = v;
      }
    }
#pragma unroll
    for (int r = 0; r < 8; ++r) {
      float acc = 0.0f;
#pragma unroll
      for (int j = 0; j < 5; ++j) acc += g_B[j] * xn[r + 4 - j];
      u[(rbase + r) * 17 + col] = acc;
    }
    // intra-wave LDS producer->consumer: split-counter wait (CDNA5)
    asm volatile("s_wait_dscnt 0" ::: "memory");

    // z = L @ u : 4 chained V_WMMA_F32_16X16X4_F32 (K=16 in K=4 slices)
    const float* Lm = (chunk == 0) ? L0sh : Lsh;
    v8f acc = {};
#pragma unroll
    for (int kk = 0; kk < 4; ++kk) {
      const int k0 = kk * 4 + half * 2;
      v2f a, b;
      a.x = Lm[col * 16 + (k0 + 0)];       // A: M=lane%16, K per 32b layout
      a.y = Lm[col * 16 + (k0 + 1)];
      b.x = u[(k0 + 0) * 17 + col];        // B: row K across lanes
      b.y = u[(k0 + 1) * 17 + col];
      acc = __builtin_amdgcn_wmma_f32_16x16x4_f32(false, a, false, b,
                                                  (short)0, acc, false, false);
    }

#pragma unroll
    for (int r = 0; r < 8; ++r)
      out[(size_t)(t0 + rbase + r) * B_COLS + c] = acc[r];
  }
}

// ---- kernel 3: sequential 4-state scan across 2048 chunks per column ----
__global__ void __launch_bounds__(32) scan_kernel(const float* __restrict__ zout,
                                                  float* __restrict__ states) {
  __shared__ float Psh[64];
  if (threadIdx.x == 0) build_P(Psh);
  __syncthreads();
  float Q[4][4];
  for (int i = 0; i < 4; ++i)
    for (int j = 0; j < 4; ++j) Q[i][j] = Psh[(12 + i) * 4 + j];

  const int c = blockIdx.x * 32 + threadIdx.x;   // column 0..1023
  float s0 = 0.0f, s1 = 0.0f, s2 = 0.0f, s3 = 0.0f;
  for (int k = 0; k < NCHUNK; ++k) {
    const size_t sb = ((size_t)k * 4) * B_COLS + c;   // state entering chunk k
    states[sb + 0 * B_COLS] = s0;
    states[sb + 1 * B_COLS] = s1;
    states[sb + 2 * B_COLS] = s2;
    states[sb + 3 * B_COLS] = s3;
    const size_t zb = ((size_t)k * CHUNK + 12) * B_COLS + c;
    // prefetch 4 chunks ahead along the serial dependence chain
    if (k + 4 < NCHUNK)
      __builtin_prefetch(&zout[zb + (size_t)(4 * CHUNK) * B_COLS], 0, 3);
    const float z0 = zout[zb + 0 * B_COLS];
    const float z1 = zout[zb + 1 * B_COLS];
    const float z2 = zout[zb + 2 * B_COLS];
    const float z3 = zout[zb + 3 * B_COLS];
    const float n0 = z0 + Q[0][0]*s0 + Q[0][1]*s1 + Q[0][2]*s2 + Q[0][3]*s3;
    const float n1 = z1 + Q[1][0]*s0 + Q[1][1]*s1 + Q[1][2]*s2 + Q[1][3]*s3;
    const float n2 = z2 + Q[2][0]*s0 + Q[2][1]*s1 + Q[2][2]*s2 + Q[2][3]*s3;
    const float n3 = z3 + Q[3][0]*s0 + Q[3][1]*s1 + Q[3][2]*s2 + Q[3][3]*s3;
    s0 = n0; s1 = n1; s2 = n2; s3 = n3;
  }
}

// ---- kernel 4: out_tile += P @ s_enter  (one K=4 WMMA per tile) ---------
__global__ void __launch_bounds__(256) fixup_kernel(const float* __restrict__ states,
                                                    float* __restrict__ out) {
  __shared__ float Psh[64];
  if (threadIdx.x == 0) build_P(Psh);
  __syncthreads();

  const int lane  = threadIdx.x & 31;
  const int wave  = threadIdx.x >> 5;
  const int col   = lane & 15;
  const int half  = lane >> 4;
  const int rbase = half * 8;

  v2f a;                                        // A = P (16x4)
  a.x = Psh[col * 4 + half * 2 + 0];
  a.y = Psh[col * 4 + half * 2 + 1];

  const int wglobal = blockIdx.x * (blockDim.x >> 5) + wave;
  const int nwaves  = gridDim.x * (blockDim.x >> 5);

  for (int tile = wglobal; tile < FIXTILES; tile += nwaves) {
    const int k     = (tile >> 6) + 1;          // chunks 1..2047
    const int ctile = tile & 63;
    const int c     = ctile * 16 + col;

    v2f b;                                      // B = state (4x16)
    const size_t sb = ((size_t)k * 4) * B_COLS + c;
    b.x = states[sb + (size_t)(half * 2 + 0) * B_COLS];
    b.y = states[sb + (size_t)(half * 2 + 1) * B_COLS];

    v8f cc;
#pragma unroll
    for (int r = 0; r < 8; ++r)
      cc[r] = out[(size_t)(k * CHUNK + rbase + r) * B_COLS + c];

    v8f d = __builtin_amdgcn_wmma_f32_16x16x4_f32(false, a, false, b,
                                                  (short)0, cc, false, false);
#pragma unroll
    for (int r = 0; r < 8; ++r)
      out[(size_t)(k * CHUNK + rbase + r) * B_COLS + c] = d[r];
  }
}

extern "C" void kernel_launch(void* const* d_in, const int* in_sizes, int n_in,
                              void* d_out, int out_size, void* d_ws, size_t ws_size,
                              hipStream_t stream) {
  (void)in_sizes; (void)n_in; (void)out_size; (void)ws_size;
  const float* sig = (const float*)d_in[0];
  const float* noi = (const float*)d_in[1];
  float* out = (float*)d_out;
  float* ws  = (float*)d_ws;
  float* sigma  = ws;            // 32768 f32 (128 KB)
  float* states = ws + T_LEN;    // 2048*4*1024 f32 (32 MB)

  sigma_kernel <<<2048, 256, 0, stream>>>(sig, sigma);
  zstate_kernel<<<2048, 256, 0, stream>>>(sig, noi, sigma, out);
  scan_kernel  <<<32,   32,  0, stream>>>(out, states);
  fixup_kernel <<<2048, 256, 0, stream>>>(states, out);
}